// CrossAggregationModuleV2_23776938951333
// MI455X (gfx1250) — compile-verified
//
#include <hip/hip_runtime.h>
#include <hip/hip_bf16.h>
#include <math.h>

typedef __attribute__((ext_vector_type(2))) float v2f;
typedef __attribute__((ext_vector_type(4))) float v4f;
typedef __attribute__((ext_vector_type(8))) float v8f;

#define NPOS 65536           // 64*64*16 spatial positions
#define QSCALE 0.35355339059327373f  // 8^-0.5

// ---- f32 WMMA: D = A(16x4) * B(4x16) + C(16x16) ----
// A layout: lane<16 -> M=lane, {a.x,a.y} = A[M][k0+0..1]; lane>=16 -> M=lane-16, K=k0+2..3
// B layout: lane<16 -> {b.x,b.y} = B[k0+0..1][N=lane]; lane>=16 -> B[k0+2..3][N=lane-16]
// C/D:      lane<16 -> vgpr r = C[r][lane]; lane>=16 -> C[8+r][lane-16]
__device__ __forceinline__ v8f wmma4(v2f a, v2f b, v8f c) {
    return __builtin_amdgcn_wmma_f32_16x16x4_f32(false, a, false, b, (short)0, c,
                                                 false, false);
}

// W is stored in LDS pair-interleaved so a B fragment is ONE aligned ds_load_b64:
//   sW[(k>>1)*128 + n*2 + (k&1)] = W[k][n]   (k even pairs; ca below is always even)
__device__ __forceinline__ v2f ldsB(const float* sWi, int ca, int n) {
    return *(const v2f*)&sWi[(ca >> 1) * 128 + n * 2];
}

// ---------------- stats zero ----------------
__global__ void zero_stats_kernel(float* __restrict__ s) {
    int t = blockIdx.x * blockDim.x + threadIdx.x;
    if (t < 512) s[t] = 0.0f;
}

// ---------------- trilinear upsample + channel stats ----------------
// x: (64, 32, 32, 8) channel-first.  xu out: [c*NPOS + p], p = (jx*64+jy)*16+jz
__global__ __launch_bounds__(256) void upsample_stats_kernel(
    const float* __restrict__ x, float* __restrict__ xu, float* __restrict__ sums) {
    int c   = blockIdx.y;
    int tid = threadIdx.x;
    int p   = blockIdx.x * 256 + tid;
    int jz = p & 15, jy = (p >> 4) & 63, jx = p >> 10;

    float px = (float)jx * (31.0f / 63.0f);
    int ix = (int)px; if (ix > 30) ix = 30; float wx = px - (float)ix;
    float py = (float)jy * (31.0f / 63.0f);
    int iy = (int)py; if (iy > 30) iy = 30; float wy = py - (float)iy;
    float pz = (float)jz * (7.0f / 15.0f);
    int iz = (int)pz; if (iz > 6)  iz = 6;  float wz = pz - (float)iz;

    const float* xb = x + c * 8192 + ix * 256 + iy * 8 + iz;
    float c000 = xb[0],   c001 = xb[1];
    float c010 = xb[8],   c011 = xb[9];
    float c100 = xb[256], c101 = xb[257];
    float c110 = xb[264], c111 = xb[265];
    float v00 = c000 + wz * (c001 - c000);
    float v01 = c010 + wz * (c011 - c010);
    float v10 = c100 + wz * (c101 - c100);
    float v11 = c110 + wz * (c111 - c110);
    float v0  = v00 + wy * (v01 - v00);
    float v1  = v10 + wy * (v11 - v10);
    float val = v0 + wx * (v1 - v0);

    xu[c * NPOS + p] = val;

    __shared__ float s1[256], s2[256];
    s1[tid] = val; s2[tid] = val * val;
    __syncthreads();
    for (int s = 128; s > 0; s >>= 1) {
        if (tid < s) { s1[tid] += s1[tid + s]; s2[tid] += s2[tid + s]; }
        __syncthreads();
    }
    if (tid == 0) {
        atomicAdd(&sums[c], s1[0]);        // sum_xu
        atomicAdd(&sums[64 + c], s2[0]);   // sumsq_xu
    }
}

// ---------------- skip channel stats ----------------
__global__ __launch_bounds__(256) void skip_stats_kernel(
    const float* __restrict__ skip, float* __restrict__ sums) {
    int c   = blockIdx.y;
    int tid = threadIdx.x;
    int p   = blockIdx.x * 256 + tid;
    float val = skip[c * NPOS + p];
    __shared__ float s1[256], s2[256];
    s1[tid] = val; s2[tid] = val * val;
    __syncthreads();
    for (int s = 128; s > 0; s >>= 1) {
        if (tid < s) { s1[tid] += s1[tid + s]; s2[tid] += s2[tid + s]; }
        __syncthreads();
    }
    if (tid == 0) {
        atomicAdd(&sums[128 + c], s1[0]);  // sum_sk
        atomicAdd(&sums[192 + c], s2[0]);  // sumsq_sk
    }
}

// sums[0..255] -> mean/rinv at [256..511]: mean_xu,rinv_xu,mean_sk,rinv_sk
__global__ void finalize_stats_kernel(float* __restrict__ s) {
    int t = threadIdx.x;
    if (t < 64) {
        float mean = s[t] * (1.0f / (float)NPOS);
        float var  = s[64 + t] * (1.0f / (float)NPOS) - mean * mean;
        s[256 + t] = mean;
        s[320 + t] = rsqrtf(var + 1e-5f);
    } else if (t < 128) {
        int c = t - 64;
        float mean = s[128 + c] * (1.0f / (float)NPOS);
        float var  = s[192 + c] * (1.0f / (float)NPOS) - mean * mean;
        s[384 + c] = mean;
        s[448 + c] = rsqrtf(var + 1e-5f);
    }
}

// ---------------- fused normalized Q/K/V projection (WMMA) ----------------
// q = (norm(skip) @ Wq + bq) * QSCALE ; k/v = norm(xu) @ Wk/Wv + bk/bv
// stats points at [mean_xu(64), rinv_xu(64), mean_sk(64), rinv_sk(64)]
__global__ __launch_bounds__(256) void qkv_kernel(
    const float* __restrict__ skip, const float* __restrict__ xu,
    const float* __restrict__ Wq, const float* __restrict__ bq,
    const float* __restrict__ Wk, const float* __restrict__ bk,
    const float* __restrict__ Wv, const float* __restrict__ bv,
    const float* __restrict__ stats,
    float* __restrict__ qo, float* __restrict__ ko, float* __restrict__ vo) {
    __shared__ __align__(16) float sW[3 * 4096];
    __shared__ float sB[3 * 64];
    __shared__ float sS[256];
    int tid = threadIdx.x;
    for (int i = tid; i < 4096; i += 256) {
        int k = i >> 6, n = i & 63;
        int idx = (k >> 1) * 128 + n * 2 + (k & 1);   // pair-interleaved
        sW[idx] = Wq[i]; sW[4096 + idx] = Wk[i]; sW[8192 + idx] = Wv[i];
    }
    if (tid < 64) { sB[tid] = bq[tid]; sB[64 + tid] = bk[tid]; sB[128 + tid] = bv[tid]; }
    sS[tid] = stats[tid];
    __syncthreads();

    int wave = tid >> 5, lane = tid & 31;
    int l16 = lane & 15, hi = lane >> 4, koff = hi << 1;
    int p0 = (blockIdx.x * 8 + wave) * 16;
    int prow = p0 + l16;

    // ---- Q (A = normalized skip) ----
    {
        v8f a0 = {0,0,0,0,0,0,0,0}, a1 = a0, a2 = a0, a3 = a0;
        for (int k0 = 0; k0 < 64; k0 += 4) {
            int ca = k0 + koff;
            float m0 = sS[128 + ca], r0 = sS[192 + ca];
            float m1 = sS[129 + ca], r1 = sS[193 + ca];
            v2f a;
            a.x = (skip[ca * NPOS + prow] - m0) * r0;
            a.y = (skip[(ca + 1) * NPOS + prow] - m1) * r1;
            a0 = wmma4(a, ldsB(sW, ca, l16),      a0);
            a1 = wmma4(a, ldsB(sW, ca, 16 + l16), a1);
            a2 = wmma4(a, ldsB(sW, ca, 32 + l16), a2);
            a3 = wmma4(a, ldsB(sW, ca, 48 + l16), a3);
        }
        int rowbase = p0 + (hi << 3);
        #pragma unroll
        for (int r = 0; r < 8; r++) {
            float* q0 = qo + (rowbase + r) * 64;
            q0[l16]      = (a0[r] + sB[l16])      * QSCALE;
            q0[16 + l16] = (a1[r] + sB[16 + l16]) * QSCALE;
            q0[32 + l16] = (a2[r] + sB[32 + l16]) * QSCALE;
            q0[48 + l16] = (a3[r] + sB[48 + l16]) * QSCALE;
        }
    }

    // ---- K and V (A = normalized xu, shared loads) ----
    {
        v8f k0a = {0,0,0,0,0,0,0,0}, k1a = k0a, k2a = k0a, k3a = k0a;
        v8f v0a = k0a, v1a = k0a, v2a = k0a, v3a = k0a;
        for (int kc = 0; kc < 64; kc += 4) {
            int ca = kc + koff;
            float m0 = sS[ca],     r0 = sS[64 + ca];
            float m1 = sS[ca + 1], r1 = sS[65 + ca];
            v2f a;
            a.x = (xu[ca * NPOS + prow] - m0) * r0;
            a.y = (xu[(ca + 1) * NPOS + prow] - m1) * r1;
            const float* wk0 = sW + 4096;
            const float* wv0 = sW + 8192;
            k0a = wmma4(a, ldsB(wk0, ca, l16),      k0a);
            k1a = wmma4(a, ldsB(wk0, ca, 16 + l16), k1a);
            k2a = wmma4(a, ldsB(wk0, ca, 32 + l16), k2a);
            k3a = wmma4(a, ldsB(wk0, ca, 48 + l16), k3a);
            v0a = wmma4(a, ldsB(wv0, ca, l16),      v0a);
            v1a = wmma4(a, ldsB(wv0, ca, 16 + l16), v1a);
            v2a = wmma4(a, ldsB(wv0, ca, 32 + l16), v2a);
            v3a = wmma4(a, ldsB(wv0, ca, 48 + l16), v3a);
        }
        int rowbase = p0 + (hi << 3);
        #pragma unroll
        for (int r = 0; r < 8; r++) {
            float* kp = ko + (rowbase + r) * 64;
            float* vp = vo + (rowbase + r) * 64;
            kp[l16]      = k0a[r] + sB[64 + l16];
            kp[16 + l16] = k1a[r] + sB[80 + l16];
            kp[32 + l16] = k2a[r] + sB[96 + l16];
            kp[48 + l16] = k3a[r] + sB[112 + l16];
            vp[l16]      = v0a[r] + sB[128 + l16];
            vp[16 + l16] = v1a[r] + sB[144 + l16];
            vp[32 + l16] = v2a[r] + sB[160 + l16];
            vp[48 + l16] = v3a[r] + sB[176 + l16];
        }
    }
}

// ---------------- 3x3x3 neighborhood attention ----------------
// one thread per (pos, head);  q/k/v/o layout: [pos*64 + head*8 + d]
__global__ __launch_bounds__(256) void attn_kernel(
    const float* __restrict__ q, const float* __restrict__ kk,
    const float* __restrict__ vv, const float* __restrict__ rpb,
    float* __restrict__ o) {
    int t = blockIdx.x * 256 + threadIdx.x;
    int h = t & 7;
    int pos = t >> 3;
    int z  = pos & 15;
    int yy = (pos >> 4) & 63;
    int xx = pos >> 10;
    int sx = xx - 1; if (sx < 0) sx = 0; if (sx > 61) sx = 61;
    int sy = yy - 1; if (sy < 0) sy = 0; if (sy > 61) sy = 61;
    int sz = z - 1;  if (sz < 0) sz = 0; if (sz > 13) sz = 13;
    int rx0 = sx - xx + 2, ry0 = sy - yy + 2, rz0 = sz - z + 2;

    const v4f* qp = (const v4f*)(q + pos * 64 + h * 8);
    v4f q0 = qp[0], q1 = qp[1];

    float lg[27];
    float mx = -1e30f;
    #pragma unroll
    for (int i = 0; i < 27; i++) {
        int a = i / 9, b = (i / 3) % 3, c = i % 3;
        int npos = ((sx + a) * 64 + (sy + b)) * 16 + (sz + c);
        const v4f* kp = (const v4f*)(kk + npos * 64 + h * 8);
        v4f k0 = kp[0], k1 = kp[1];
        v4f pd = q0 * k0 + q1 * k1;
        float dot = pd.x + pd.y + pd.z + pd.w;
        float bias = rpb[((h * 5 + rx0 + a) * 5 + ry0 + b) * 5 + rz0 + c];
        lg[i] = dot + bias;
        mx = fmaxf(mx, lg[i]);
    }
    float sum = 0.0f;
    #pragma unroll
    for (int i = 0; i < 27; i++) { lg[i] = __expf(lg[i] - mx); sum += lg[i]; }
    float inv = 1.0f / sum;

    v4f acc0 = {0, 0, 0, 0}, acc1 = acc0;
    #pragma unroll
    for (int i = 0; i < 27; i++) {
        int a = i / 9, b = (i / 3) % 3, c = i % 3;
        int npos = ((sx + a) * 64 + (sy + b)) * 16 + (sz + c);
        const v4f* vp = (const v4f*)(vv + npos * 64 + h * 8);
        float w = lg[i];
        acc0 = acc0 + vp[0] * w;
        acc1 = acc1 + vp[1] * w;
    }
    v4f* op = (v4f*)(o + pos * 64 + h * 8);
    op[0] = acc0 * inv;
    op[1] = acc1 * inv;
}

// ---------------- output projection (WMMA), write channel-first ----------------
// Epilogue transposes the 16x64 tile through LDS so global stores are coalesced b128.
__global__ __launch_bounds__(256) void oproj_kernel(
    const float* __restrict__ o, const float* __restrict__ Wo,
    const float* __restrict__ bo, float* __restrict__ out) {
    __shared__ __align__(16) float sW[4096];
    __shared__ float sB[64];
    __shared__ __align__(16) float sT[8 * 1024];   // per-wave 64ch x 16pos tile
    int tid = threadIdx.x;
    for (int i = tid; i < 4096; i += 256) {
        int k = i >> 6, n = i & 63;
        sW[(k >> 1) * 128 + n * 2 + (k & 1)] = Wo[i];
    }
    if (tid < 64) sB[tid] = bo[tid];
    __syncthreads();

    int wave = tid >> 5, lane = tid & 31;
    int l16 = lane & 15, hi = lane >> 4, koff = hi << 1;
    int p0 = (blockIdx.x * 8 + wave) * 16;
    const float* arow = o + (p0 + l16) * 64;

    v8f a0 = {0,0,0,0,0,0,0,0}, a1 = a0, a2 = a0, a3 = a0;
    for (int k0 = 0; k0 < 64; k0 += 4) {
        int ca = k0 + koff;
        v2f a = *(const v2f*)&arow[ca];     // 8B-aligned (ca even)
        a0 = wmma4(a, ldsB(sW, ca, l16),      a0);
        a1 = wmma4(a, ldsB(sW, ca, 16 + l16), a1);
        a2 = wmma4(a, ldsB(sW, ca, 32 + l16), a2);
        a3 = wmma4(a, ldsB(sW, ca, 48 + l16), a3);
    }

    // transpose tile to [channel][pos] in LDS, adding bias
    float* tw = &sT[wave * 1024];
    #pragma unroll
    for (int r = 0; r < 8; r++) {
        int lp = (hi << 3) + r;             // local position 0..15
        tw[l16 * 16 + lp]        = a0[r] + sB[l16];
        tw[(16 + l16) * 16 + lp] = a1[r] + sB[16 + l16];
        tw[(32 + l16) * 16 + lp] = a2[r] + sB[32 + l16];
        tw[(48 + l16) * 16 + lp] = a3[r] + sB[48 + l16];
    }
    __syncthreads();

    // coalesced writes: 4 lanes cover one channel's 16 floats (4x b128)
    #pragma unroll
    for (int it = 0; it < 8; it++) {
        int c = (lane >> 2) + it * 8;
        int j = (lane & 3) << 2;
        v4f val = *(const v4f*)&tw[c * 16 + j];
        *(v4f*)&out[c * NPOS + p0 + j] = val;
    }
}

extern "C" void kernel_launch(void* const* d_in, const int* in_sizes, int n_in,
                              void* d_out, int out_size, void* d_ws, size_t ws_size,
                              hipStream_t stream) {
    const float* x    = (const float*)d_in[0];
    const float* skip = (const float*)d_in[1];
    const float* Wq   = (const float*)d_in[2];
    const float* bq   = (const float*)d_in[3];
    const float* Wk   = (const float*)d_in[4];
    const float* bk   = (const float*)d_in[5];
    const float* Wv   = (const float*)d_in[6];
    const float* bv   = (const float*)d_in[7];
    const float* Wo   = (const float*)d_in[8];
    const float* bo   = (const float*)d_in[9];
    const float* rpb  = (const float*)d_in[10];
    float* out = (float*)d_out;

    float* ws = (float*)d_ws;
    float* XU = ws;                       // 64*NPOS
    float* Q  = ws + 4194304;             // NPOS*64
    float* Kb = ws + 8388608;
    float* Vb = ws + 12582912;
    float* O  = ws + 16777216;
    float* S  = ws + 20971520;            // 512 floats: sums(256) + mean/rinv(256)

    zero_stats_kernel<<<2, 256, 0, stream>>>(S);
    {
        dim3 grid(NPOS / 256, 64);
        upsample_stats_kernel<<<grid, 256, 0, stream>>>(x, XU, S);
        skip_stats_kernel<<<grid, 256, 0, stream>>>(skip, S);
    }
    finalize_stats_kernel<<<1, 128, 0, stream>>>(S);
    qkv_kernel<<<NPOS / 16 / 8, 256, 0, stream>>>(skip, XU, Wq, bq, Wk, bk, Wv, bv,
                                                  S + 256, Q, Kb, Vb);
    attn_kernel<<<NPOS * 8 / 256, 256, 0, stream>>>(Q, Kb, Vb, rpb, O);
    oproj_kernel<<<NPOS / 16 / 8, 256, 0, stream>>>(O, Wo, bo, out);
}